// SpatialTemporalSynchronousAttention_30528627540107
// MI455X (gfx1250) — compile-verified
//
#include <hip/hip_runtime.h>
#include <math.h>

#define Bz 32
#define Nn 510
#define Dm 64
#define Hh 4
#define HD (Hh*Dm)          // 256
#define ALPHA 0.1f
#define LEAKY 0.01f
#define LN_EPS 1e-5f

typedef __attribute__((ext_vector_type(2))) float v2f;
typedef __attribute__((ext_vector_type(8))) float v8f;

#ifndef __has_builtin
#define __has_builtin(x) 0
#endif

#if defined(__gfx1250__) && __has_builtin(__builtin_amdgcn_wmma_f32_16x16x4_f32)
#define USE_WMMA_F32 1
#endif

// D(16x16,f32) = A(16x4,f32) x B(4x16,f32) + C  — native fp32 WMMA on CDNA5.
__device__ __forceinline__ v8f wmma_16x16x4_f32(v2f a, v2f b, v8f c) {
#ifdef USE_WMMA_F32
  return __builtin_amdgcn_wmma_f32_16x16x4_f32(false, a, false, b, (short)0, c,
                                               false, false);
#else
  c[0] += a[0] * b[0] + a[1] * b[1];  // compile-safety fallback only
  return c;
#endif
}

// ---------------------------------------------------------------------------
// K1: xt = x @ W   (M=16320, N=256, K=64).
// One wave computes a 16x64 strip (4 accumulators) so each A fragment of x is
// loaded once and feeds 4 WMMAs -> x traffic cut 4x vs tile-per-wave.
// ---------------------------------------------------------------------------
__global__ __launch_bounds__(256) void k_xt_gemm(const float* __restrict__ x,
                                                 const float* __restrict__ W,
                                                 float* __restrict__ xt) {
  const int wave = blockIdx.x * (blockDim.x >> 5) + (threadIdx.x >> 5);
  const int lane = threadIdx.x & 31;
  const int mt = wave >> 2;        // 0..1019  (16-row strip)
  const int ng = wave & 3;         // 0..3     (64-col group)
  const int half = lane >> 4;
  const int ll = lane & 15;
  const int arow = mt * 16 + ll;   // always < 16320
  const int colb = ng * 64 + ll;
  v8f acc[4];
  #pragma unroll
  for (int nt = 0; nt < 4; ++nt) acc[nt] = (v8f){0.f,0.f,0.f,0.f,0.f,0.f,0.f,0.f};
  const float* xr = x + (size_t)arow * Dm;
  for (int k0 = 0; k0 < Dm; k0 += 4) {
    const int kb = k0 + 2 * half;  // A layout: lanes 0-15 -> K{0,1}, 16-31 -> K{2,3}
    v2f a;  a[0] = xr[kb];  a[1] = xr[kb + 1];
    #pragma unroll
    for (int nt = 0; nt < 4; ++nt) {
      v2f bf;
      bf[0] = W[kb * HD + colb + nt * 16];
      bf[1] = W[(kb + 1) * HD + colb + nt * 16];
      acc[nt] = wmma_16x16x4_f32(a, bf, acc[nt]);
    }
  }
  #pragma unroll
  for (int nt = 0; nt < 4; ++nt)
    for (int v = 0; v < 8; ++v) {
      const int row = mt * 16 + v + 8 * half;  // C/D: VGPR v holds rows v, v+8
      xt[(size_t)row * HD + colb + nt * 16] = acc[nt][v];
    }
}

// ---------------------------------------------------------------------------
// K2: s_i[b,h,n] = <xt[b,n,h,:], w1[h]>, s_j likewise with w2
// ---------------------------------------------------------------------------
__global__ void k_scores(const float* __restrict__ xt,
                         const float* __restrict__ aw,
                         float* __restrict__ si, float* __restrict__ sj) {
  const int idx = blockIdx.x * blockDim.x + threadIdx.x;
  if (idx >= Bz * Hh * Nn) return;
  const int n = idx % Nn;
  const int h = (idx / Nn) % Hh;
  const int b = idx / (Nn * Hh);
  const float* row = xt + ((size_t)(b * Nn + n)) * HD + h * Dm;
  const float* w1 = aw + h * 2 * Dm;
  const float* w2 = w1 + Dm;
  float a1 = 0.f, a2 = 0.f;
  for (int d = 0; d < Dm; ++d) {
    const float v = row[d];
    a1 += v * w1[d];
    a2 += v * w2[d];
  }
  si[idx] = a1;
  sj[idx] = a2;
}

// ---------------------------------------------------------------------------
// K3: fused leaky -> head-mean -> 3x3 conv -> alpha-blend -> softmax.
// One 512-thread block per (b, i); A_eva/A_mean/A_st never touch HBM.
// Softmax reductions: wave32 shfl_xor butterflies + 16-entry LDS partials
// (3 block barriers per head instead of ~20).
// ---------------------------------------------------------------------------
__global__ __launch_bounds__(512) void k_attn_rows(
    const float* __restrict__ si, const float* __restrict__ sj,
    const float* __restrict__ causal, const float* __restrict__ cw,
    const float* __restrict__ cb, float* __restrict__ Aout) {
  __shared__ float sjs[Hh][512];
  __shared__ float amean[3][514];  // +2 halo columns
  __shared__ float cg[3][514];
  __shared__ float wredm[16];
  __shared__ float wreds[16];
  const int b = blockIdx.x / Nn;
  const int i = blockIdx.x % Nn;
  const int j = threadIdx.x;
  const bool jv = (j < Nn);

  for (int h = 0; h < Hh; ++h)
    sjs[h][j] = jv ? sj[(b * Hh + h) * Nn + j] : 0.f;
  __syncthreads();

  for (int r3 = 0; r3 < 3; ++r3) {
    const int r = i - 1 + r3;
    float am = 0.f, cgv = 0.f;
    if (r >= 0 && r < Nn && jv) {
      for (int h = 0; h < Hh; ++h) {
        const float e = si[(b * Hh + h) * Nn + r] + sjs[h][j];
        am += (e >= 0.f) ? e : LEAKY * e;
      }
      am *= 0.25f;
      cgv = causal[(size_t)r * Nn + j];
    }
    amean[r3][j + 1] = am;
    cg[r3][j + 1] = cgv;
    if (j == 0) {
      amean[r3][0] = 0.f;   cg[r3][0] = 0.f;
      amean[r3][513] = 0.f; cg[r3][513] = 0.f;
    }
  }
  __syncthreads();

  for (int h = 0; h < Hh; ++h) {
    float conv = cb[h];
    #pragma unroll
    for (int dr = 0; dr < 3; ++dr)
      #pragma unroll
      for (int dc = 0; dc < 3; ++dc) {
        conv += cw[((h * 2 + 0) * 3 + dr) * 3 + dc] * amean[dr][j + dc];
        conv += cw[((h * 2 + 1) * 3 + dr) * 3 + dc] * cg[dr][j + dc];
      }
    const float e = si[(b * Hh + h) * Nn + i] + sjs[h][j];
    const float aeva = (e >= 0.f) ? e : LEAKY * e;
    const float logit = jv ? (ALPHA * aeva + (1.f - ALPHA) * conv) : -3.0e38f;

    // --- block max: wave32 butterfly, then 16 partials in LDS ---
    float wm = logit;
    #pragma unroll
    for (int off = 16; off > 0; off >>= 1)
      wm = fmaxf(wm, __shfl_xor(wm, off, 32));
    if ((j & 31) == 0) wredm[j >> 5] = wm;
    __syncthreads();
    float m = wredm[0];
    #pragma unroll
    for (int w = 1; w < 16; ++w) m = fmaxf(m, wredm[w]);

    const float ex = jv ? __expf(logit - m) : 0.f;

    // --- block sum ---
    float ws = ex;
    #pragma unroll
    for (int off = 16; off > 0; off >>= 1)
      ws += __shfl_xor(ws, off, 32);
    if ((j & 31) == 0) wreds[j >> 5] = ws;
    __syncthreads();
    float denom = 0.f;
    #pragma unroll
    for (int w = 0; w < 16; ++w) denom += wreds[w];

    if (jv)
      Aout[((size_t)(b * Hh + h) * Nn + i) * Nn + j] = ex / denom;
    __syncthreads();  // protect wredm/wreds before next head
  }
}

// ---------------------------------------------------------------------------
// K4: out[b,n,d] = 0.25 * sum_h  A[b,h] (510x510) @ xt[b,:,h,:] (510x64)
// One wave owns a full 16x64 strip: each A fragment is loaded from HBM once
// and feeds 4 WMMAs (A is the 133 MB tensor -> this is the traffic that
// matters at 23.3 TB/s). Main loop is maskless; single masked tail step
// handles K=510 % 4. global_prefetch on the streamed A rows.
// ---------------------------------------------------------------------------
__global__ __launch_bounds__(256) void k_attn_apply(const float* __restrict__ A,
                                                    const float* __restrict__ xt,
                                                    float* __restrict__ outb) {
  const int wave = blockIdx.x * (blockDim.x >> 5) + (threadIdx.x >> 5);
  const int lane = threadIdx.x & 31;
  const int b  = wave >> 5;        // 0..31
  const int mt = wave & 31;        // 0..31 (16-row strip; last strip partial)
  const int half = lane >> 4;
  const int ll = lane & 15;
  const int arow = mt * 16 + ll;
  const bool rv = (arow < Nn);
  v8f acc[4];
  #pragma unroll
  for (int nt = 0; nt < 4; ++nt) acc[nt] = (v8f){0.f,0.f,0.f,0.f,0.f,0.f,0.f,0.f};

  for (int h = 0; h < Hh; ++h) {
    const float* Ar = A + ((size_t)(b * Hh + h) * Nn + (rv ? arow : 0)) * Nn;
    const float* Xh = xt + (size_t)b * Nn * HD + h * Dm + ll;
    // main loop: k in [0, 508), no masking needed (kb max = 506, +1 = 507)
    for (int k0 = 0; k0 < 508; k0 += 4) {
      const int kb = k0 + 2 * half;
      if ((k0 & 31) == 0) __builtin_prefetch(Ar + kb + 32, 0, 3);
      v2f a;
      a[0] = rv ? Ar[kb]     : 0.f;
      a[1] = rv ? Ar[kb + 1] : 0.f;
      #pragma unroll
      for (int nt = 0; nt < 4; ++nt) {
        v2f bf;
        bf[0] = Xh[(size_t)kb * HD + nt * 16];
        bf[1] = Xh[(size_t)(kb + 1) * HD + nt * 16];
        acc[nt] = wmma_16x16x4_f32(a, bf, acc[nt]);
      }
    }
    // tail: k0 = 508 -> half 0 covers k=508,509; half 1 lanes contribute 0
    {
      const int kb = 508 + 2 * half;
      const int kc0 = (kb < Nn) ? kb : (Nn - 1);
      const int kc1 = (kb + 1 < Nn) ? (kb + 1) : (Nn - 1);
      v2f a;
      a[0] = (rv && kb < Nn)     ? Ar[kc0] : 0.f;
      a[1] = (rv && kb + 1 < Nn) ? Ar[kc1] : 0.f;
      #pragma unroll
      for (int nt = 0; nt < 4; ++nt) {
        v2f bf;
        const float b0 = Xh[(size_t)kc0 * HD + nt * 16];
        const float b1 = Xh[(size_t)kc1 * HD + nt * 16];
        bf[0] = (kb < Nn)     ? b0 : 0.f;
        bf[1] = (kb + 1 < Nn) ? b1 : 0.f;
        acc[nt] = wmma_16x16x4_f32(a, bf, acc[nt]);
      }
    }
  }
  #pragma unroll
  for (int nt = 0; nt < 4; ++nt)
    for (int v = 0; v < 8; ++v) {
      const int row = mt * 16 + v + 8 * half;
      if (row < Nn)
        outb[((size_t)b * Nn + row) * Dm + nt * 16 + ll] = 0.25f * acc[nt][v];
    }
}

// ---------------------------------------------------------------------------
// K5: g = out @ fcg_w + fcg_b ; GLU ; residual ; LayerNorm. Block per (b,n).
// ---------------------------------------------------------------------------
__global__ __launch_bounds__(128) void k_final(const float* __restrict__ outb,
                                               const float* __restrict__ x,
                                               const float* __restrict__ fw,
                                               const float* __restrict__ fb,
                                               const float* __restrict__ lg,
                                               const float* __restrict__ lb,
                                               float* __restrict__ y) {
  __shared__ float orow[Dm];
  __shared__ float garr[2 * Dm];
  __shared__ float red[Dm];
  const int bn = blockIdx.x;  // b*Nn + n
  const int t = threadIdx.x;  // 0..127
  if (t < Dm) orow[t] = outb[(size_t)bn * Dm + t];
  __syncthreads();
  float g = fb[t];
  for (int d = 0; d < Dm; ++d) g += orow[d] * fw[d * 2 * Dm + t];
  garr[t] = g;
  __syncthreads();
  float yv = 0.f;
  if (t < Dm) {
    const float a = garr[t];
    const float bg = garr[Dm + t];
    const float glu = a * (1.f / (1.f + __expf(-bg)));
    yv = glu + x[(size_t)bn * Dm + t];
    red[t] = yv;
  }
  __syncthreads();
  for (int s = 32; s > 0; s >>= 1) {
    if (t < s) red[t] += red[t + s];
    __syncthreads();
  }
  const float mu = red[0] * (1.f / Dm);
  __syncthreads();
  if (t < Dm) {
    const float d0 = yv - mu;
    red[t] = d0 * d0;
  }
  __syncthreads();
  for (int s = 32; s > 0; s >>= 1) {
    if (t < s) red[t] += red[t + s];
    __syncthreads();
  }
  const float var = red[0] * (1.f / Dm);
  if (t < Dm) {
    const float norm = (yv - mu) * rsqrtf(var + LN_EPS);
    y[(size_t)bn * Dm + t] = norm * lg[t] + lb[t];
  }
}

// ---------------------------------------------------------------------------
extern "C" void kernel_launch(void* const* d_in, const int* in_sizes, int n_in,
                              void* d_out, int out_size, void* d_ws,
                              size_t ws_size, hipStream_t stream) {
  const float* x   = (const float*)d_in[0];
  const float* cgp = (const float*)d_in[1];
  const float* W   = (const float*)d_in[2];
  const float* aw  = (const float*)d_in[3];
  const float* cw  = (const float*)d_in[4];
  const float* cb  = (const float*)d_in[5];
  const float* fw  = (const float*)d_in[6];
  const float* fb  = (const float*)d_in[7];
  const float* lg  = (const float*)d_in[8];
  const float* lb  = (const float*)d_in[9];

  float* out = (float*)d_out;
  float* yout = out;                           // (32,510,64)
  float* Aout = out + (size_t)Bz * Nn * Dm;    // (32,4,510,510)

  float* ws = (float*)d_ws;
  size_t o = 0;
  float* xt = ws + o; o += (size_t)Bz * Nn * HD;   // 4,177,920 f
  float* si = ws + o; o += (size_t)Bz * Hh * Nn;   //    65,280 f
  float* sj = ws + o; o += (size_t)Bz * Hh * Nn;   //    65,280 f
  float* ob = ws + o;                               // 1,044,480 f  (~21.4 MB total)

  // K1: 1020 strips x 4 col groups = 4080 waves, 8 waves/block
  k_xt_gemm<<<510, 256, 0, stream>>>(x, W, xt);

  const int nscore = Bz * Hh * Nn;
  k_scores<<<(nscore + 255) / 256, 256, 0, stream>>>(xt, aw, si, sj);

  k_attn_rows<<<Bz * Nn, 512, 0, stream>>>(si, sj, cgp, cw, cb, Aout);

  // K4: 32 b x 32 strips = 1024 waves, 8 waves/block
  k_attn_apply<<<128, 256, 0, stream>>>(Aout, xt, ob);

  k_final<<<Bz * Nn, 128, 0, stream>>>(ob, x, fw, fb, lg, lb, yout);
}